// DyadicMambaScan_7249904795698
// MI455X (gfx1250) — compile-verified
//
#include <hip/hip_runtime.h>

// Problem shape (fixed by the reference).
#define BDIM   8
#define SDIM   4096
#define DDIM   1024

// Tiling.
#define TILE   256                 // channels per block == threads per block (8 wave32)
#define CHUNK  128                 // timesteps per chunk
#define NCHUNK (SDIM / CHUNK)      // 32
#define STEPS  8                   // timesteps staged per async stage
#define NSTAGE (CHUNK / STEPS)     // 16

// LDS layout: double buffer x 3 arrays x (STEPS*TILE) f32/i32.
#define STAGE_BYTES (STEPS * TILE * 4)   // 8192
#define BUF_BYTES   (3 * STAGE_BYTES)    // 24576
#define LDS_BYTES   (2 * BUF_BYTES)      // 49152

typedef unsigned u32x4 __attribute__((ext_vector_type(4)));
typedef unsigned u32x8 __attribute__((ext_vector_type(8)));

// ---------------------------------------------------------------------------
// CDNA5 async global -> LDS copy (VGLOBAL GVS form: vdst = LDS byte addr,
// voff = 32-bit byte offset, saddr = 64-bit base). Tracked by ASYNCcnt.
// ---------------------------------------------------------------------------
__device__ __forceinline__ void async_ld_b128(unsigned ldsAddr, unsigned gOff,
                                              const void* sBase) {
    asm volatile("global_load_async_to_lds_b128 %0, %1, %2"
                 :
                 : "v"(ldsAddr), "v"(gOff), "s"(sBase)
                 : "memory");
}

__device__ __forceinline__ void wait_async_le6() {
    asm volatile("s_wait_asynccnt 0x6" ::: "memory");
}
__device__ __forceinline__ void wait_async_0() {
    asm volatile("s_wait_asynccnt 0x0" ::: "memory");
}

// ---------------------------------------------------------------------------
// CDNA5 Tensor Data Mover: 2D tile load (STEPS x TILE f32, row stride DDIM)
// from global to LDS. D# built per ISA ch.8 (group0 128b + group1 256b),
// 2-group form for <=2D tensors. Tracked by TENSORcnt; EXEC ignored, so it
// must be issued by exactly one wave per workgroup.
// ---------------------------------------------------------------------------
__device__ __forceinline__ void tdm_load_2d(const void* gTileStart,
                                            unsigned ldsAddr) {
    const unsigned long long ga = (unsigned long long)gTileStart;
    u32x4 g0;
    g0[0] = 0x1u;                                   // count=1, user D#
    g0[1] = ldsAddr;                                // lds_addr (bytes)
    g0[2] = (unsigned)(ga & 0xffffffffu);           // global_addr[31:0]
    g0[3] = (unsigned)((ga >> 32) & 0x01ffffffu)    // global_addr[56:32]
          | (2u << 30);                             // type=2 ("image")
    u32x8 g1;
    g1[0] = (2u << 16);                 // workgroup_mask=0, data_size=2 (4B)
    g1[1] = ((unsigned)TILE & 0xffffu) << 16;           // tensor_dim0[15:0]
    g1[2] = ((unsigned)TILE >> 16)                      // tensor_dim0[31:16]
          | (((unsigned)STEPS & 0xffffu) << 16);        // tensor_dim1[15:0]
    g1[3] = ((unsigned)STEPS >> 16)                     // tensor_dim1[31:16]
          | (((unsigned)TILE & 0xffffu) << 16);         // tile_dim0 = TILE
    g1[4] = (unsigned)STEPS & 0xffffu;                  // tile_dim1 = STEPS
    g1[5] = (unsigned)DDIM;                             // dim0_stride[31:0]
    g1[6] = 0u;                                         // dim0_stride[47:32]
    g1[7] = 0u;                                         // dim1_stride (unused)
    asm volatile("tensor_load_to_lds %0, %1"
                 :
                 : "s"(g0), "s"(g1)
                 : "memory");
}

__device__ __forceinline__ void tdm_issue_stage(const char* xB, const char* nB,
                                                const char* sB,
                                                unsigned baseByte, int t0s,
                                                int buf) {
    const unsigned ldsBase = (unsigned)buf * BUF_BYTES;
    const unsigned gOff    = baseByte + (unsigned)t0s * (DDIM * 4u);
    tdm_load_2d(xB + gOff, ldsBase + 0u * STAGE_BYTES);
    tdm_load_2d(nB + gOff, ldsBase + 1u * STAGE_BYTES);
    tdm_load_2d(sB + gOff, ldsBase + 2u * STAGE_BYTES);
}

// Issue one stage via per-lane async copies: STEPS rows x TILE cols of
// x/num/shift. 8 KB per array => 512 b128 granules => 2/thread/array.
__device__ __forceinline__ void issue_stage(const char* xB, const char* nB,
                                            const char* sB, unsigned baseByte,
                                            int t0s, int tid, int buf) {
    const unsigned ldsBase = (unsigned)buf * BUF_BYTES;
#pragma unroll
    for (int k = 0; k < 2; ++k) {
        const unsigned q    = (unsigned)tid + (unsigned)k * TILE;  // 0..511
        const unsigned row  = q >> 6;                // 64 x 16B granules/row
        const unsigned colb = (q & 63u) * 16u;
        const unsigned goff =
            baseByte + (unsigned)(t0s + (int)row) * (DDIM * 4u) + colb;
        const unsigned l = ldsBase + q * 16u;
        async_ld_b128(l + 0u * STAGE_BYTES, goff, xB);
        async_ld_b128(l + 1u * STAGE_BYTES, goff, nB);
        async_ld_b128(l + 2u * STAGE_BYTES, goff, sB);
    }
}

// ---------------------------------------------------------------------------
// Phase 1: per (b, chunk, channel-tile) compute the chunk's affine aggregate
// (A, B): h_out = A * h_in + B.  Staging: GLOBAL_LOAD_ASYNC_TO_LDS (ASYNCcnt).
// ---------------------------------------------------------------------------
__global__ __launch_bounds__(TILE) void scan_phase1(
    const float* __restrict__ x, const int* __restrict__ dn,
    const int* __restrict__ dsh, float* __restrict__ Aagg,
    float* __restrict__ Bagg) {
    extern __shared__ char smem[];
    const int tid = threadIdx.x;
    const int wg  = blockIdx.x;                 // B * NCHUNK * (D/TILE) = 1024
    const int jt  = wg & 3;                     // D/TILE = 4
    const int c   = (wg >> 2) & (NCHUNK - 1);   // 32
    const int b   = wg >> 7;
    const int tile_base = jt * TILE;
    const int t0        = c * CHUNK;
    const unsigned baseByte =
        ((unsigned)b * SDIM * DDIM + (unsigned)tile_base) * 4u;

    float A = 1.0f, Bv = 0.0f;

    issue_stage((const char*)x, (const char*)dn, (const char*)dsh, baseByte,
                t0, tid, 0);

    for (int st = 0; st < NSTAGE; ++st) {
        if (st + 1 < NSTAGE) {
            issue_stage((const char*)x, (const char*)dn, (const char*)dsh,
                        baseByte, t0 + (st + 1) * STEPS, tid, (st + 1) & 1);
            wait_async_le6();   // current stage's 6 copies retired (in-order)
        } else {
            wait_async_0();
        }
        __syncthreads();

        const char*  bb = smem + (st & 1) * BUF_BYTES;
        const float* xs = (const float*)(bb + 0 * STAGE_BYTES);
        const int*   ns = (const int*)(bb + 1 * STAGE_BYTES);
        const int*   ss = (const int*)(bb + 2 * STAGE_BYTES);
#pragma unroll
        for (int k = 0; k < STEPS; ++k) {
            const float xv = xs[k * TILE + tid];
            const float f  = ldexpf((float)ns[k * TILE + tid],
                                    -ss[k * TILE + tid]);  // v_ldexp_f32, exact
            A  = A * f;
            Bv = fmaf(Bv, f, xv);
        }
        __syncthreads();  // protect buf (st&1) before stage st+2 overwrites it
    }

    const unsigned idx =
        ((unsigned)b * NCHUNK + (unsigned)c) * DDIM + (unsigned)tile_base + tid;
    Aagg[idx] = A;
    Bagg[idx] = Bv;
}

// ---------------------------------------------------------------------------
// Phase 2: exclusive scan of chunk aggregates along the 32 chunks per (b,d).
// ---------------------------------------------------------------------------
__global__ __launch_bounds__(256) void scan_phase2(
    const float* __restrict__ Aagg, const float* __restrict__ Bagg,
    float* __restrict__ Hin) {
    const int gid = blockIdx.x * blockDim.x + threadIdx.x;  // 0..B*D-1
    const int b   = gid >> 10;       // / DDIM
    const int d   = gid & (DDIM - 1);
    float h = 0.0f;
#pragma unroll 4
    for (int c = 0; c < NCHUNK; ++c) {
        const unsigned idx = ((unsigned)b * NCHUNK + (unsigned)c) * DDIM + d;
        Hin[idx] = h;                       // exclusive prefix (chunk entry h)
        h = fmaf(Aagg[idx], h, Bagg[idx]);
    }
}

// ---------------------------------------------------------------------------
// Phase 3: replay each chunk seeded with its entry state; stream outputs.
// Staging: TENSOR_LOAD_TO_LDS (TDM, TENSORcnt), wave 0 is the sole issuer.
// ---------------------------------------------------------------------------
__global__ __launch_bounds__(TILE) void scan_phase3(
    const float* __restrict__ x, const int* __restrict__ dn,
    const int* __restrict__ dsh, const float* __restrict__ Hin,
    float* __restrict__ out) {
    extern __shared__ char smem[];
    const int tid = threadIdx.x;
    const int wg  = blockIdx.x;
    const int jt  = wg & 3;
    const int c   = (wg >> 2) & (NCHUNK - 1);
    const int b   = wg >> 7;
    const int tile_base = jt * TILE;
    const int t0        = c * CHUNK;
    const unsigned baseByte =
        ((unsigned)b * SDIM * DDIM + (unsigned)tile_base) * 4u;

    const unsigned aidx =
        ((unsigned)b * NCHUNK + (unsigned)c) * DDIM + (unsigned)tile_base + tid;
    float h = Hin[aidx];

    const bool issuer = (tid < 32);   // wave 0 only: TDM ignores EXEC

    if (issuer)
        tdm_issue_stage((const char*)x, (const char*)dn, (const char*)dsh,
                        baseByte, t0, 0);

    for (int st = 0; st < NSTAGE; ++st) {
        if (st + 1 < NSTAGE) {
            if (issuer) {
                tdm_issue_stage((const char*)x, (const char*)dn,
                                (const char*)dsh, baseByte,
                                t0 + (st + 1) * STEPS, (st + 1) & 1);
                // 3 current + 3 next outstanding; <=3 means current stage done
                __builtin_amdgcn_s_wait_tensorcnt(3);
            }
        } else {
            if (issuer) __builtin_amdgcn_s_wait_tensorcnt(0);
        }
        __syncthreads();   // publish stage st LDS data to all waves

        const char*  bb = smem + (st & 1) * BUF_BYTES;
        const float* xs = (const float*)(bb + 0 * STAGE_BYTES);
        const int*   ns = (const int*)(bb + 1 * STAGE_BYTES);
        const int*   ss = (const int*)(bb + 2 * STAGE_BYTES);
#pragma unroll
        for (int k = 0; k < STEPS; ++k) {
            const float xv = xs[k * TILE + tid];
            const float f  = ldexpf((float)ns[k * TILE + tid],
                                    -ss[k * TILE + tid]);
            h = fmaf(h, f, xv);
            const int t = t0 + st * STEPS + k;
            out[(unsigned)(b * SDIM + t) * DDIM + (unsigned)tile_base + tid] = h;
        }
        __syncthreads();  // all waves done with buf (st&1) before reuse
    }
}

extern "C" void kernel_launch(void* const* d_in, const int* in_sizes, int n_in,
                              void* d_out, int out_size, void* d_ws,
                              size_t ws_size, hipStream_t stream) {
    const float* x   = (const float*)d_in[0];
    const int*   dn  = (const int*)d_in[1];
    const int*   dsh = (const int*)d_in[2];
    float*       out = (float*)d_out;

    const size_t aggN = (size_t)BDIM * NCHUNK * DDIM;  // 256K floats = 1 MB
    float* Aagg = (float*)d_ws;
    float* Bagg = Aagg + aggN;
    float* Hin  = Bagg + aggN;   // needs 3 MB of d_ws total

    const int nBlocks = BDIM * NCHUNK * (DDIM / TILE);  // 1024
    scan_phase1<<<nBlocks, TILE, LDS_BYTES, stream>>>(x, dn, dsh, Aagg, Bagg);
    scan_phase2<<<(BDIM * DDIM) / 256, 256, 0, stream>>>(Aagg, Bagg, Hin);
    scan_phase3<<<nBlocks, TILE, LDS_BYTES, stream>>>(x, dn, dsh, Hin, out);
}